// SnakeFastLoss_8546984919403
// MI455X (gfx1250) — compile-verified
//
#include <hip/hip_runtime.h>
#include <math.h>

// ---- problem constants (match reference) ----
constexpr int kB  = 8;
constexpr int kH  = 384;
constexpr int kW  = 384;
constexpr int kHW = kH * kW;          // 147456
constexpr int kN  = 96;               // nodes
constexpr float kStep  = 0.1f;
constexpr float kAlpha = 0.01f;
constexpr float kBeta  = 0.01f;
constexpr int   kSteps  = 50;
constexpr int   kStepsW = 10;
constexpr float kDmax = 15.0f;
constexpr float kExtF = 10.0f;

typedef __attribute__((ext_vector_type(2))) float v2f;
typedef __attribute__((ext_vector_type(8))) float v8f;

// ---------------- bilinear interp with clamping (matches reference interp) ----------------
__device__ __forceinline__ float bilerp(const float* __restrict__ img, float y, float x) {
    y = fminf(fmaxf(y, 0.f), (float)(kH - 1));
    x = fminf(fmaxf(x, 0.f), (float)(kW - 1));
    int y0 = (int)floorf(y);
    int x0 = (int)floorf(x);
    int y1 = y0 + 1 < kH ? y0 + 1 : kH - 1;
    int x1 = x0 + 1 < kW ? x0 + 1 : kW - 1;
    float wy = y - (float)y0;
    float wx = x - (float)x0;
    float v00 = img[y0 * kW + x0];
    float v01 = img[y0 * kW + x1];
    float v10 = img[y1 * kW + x0];
    float v11 = img[y1 * kW + x1];
    return v00 * (1.f - wy) * (1.f - wx) + v01 * (1.f - wy) * wx +
           v10 * wy * (1.f - wx) + v11 * wy * wx;
}

// ---------------- 7x7 SAME cross-correlation, 2 channels, raw + abs ----------------
__global__ void conv_kernel(const float* __restrict__ pred, const float* __restrict__ fltr,
                            float* __restrict__ gimg, float* __restrict__ gimgW) {
    __shared__ float sf[98];
    int t = threadIdx.x;
    if (t < 98) sf[t] = fltr[t];
    __syncthreads();

    long gid = (long)blockIdx.x * blockDim.x + threadIdx.x;
    if (gid >= (long)kB * kHW) return;
    int b = (int)(gid / kHW);
    int p = (int)(gid - (long)b * kHW);
    int y = p / kW;
    int x = p - y * kW;
    const float* in = pred + (long)b * kHW;

    float a0 = 0.f, a1 = 0.f, w0 = 0.f, w1 = 0.f;
    #pragma unroll
    for (int ky = 0; ky < 7; ky++) {
        int iy = y + ky - 3;
        if (iy < 0 || iy >= kH) continue;
        #pragma unroll
        for (int kx = 0; kx < 7; kx++) {
            int ix = x + kx - 3;
            if (ix < 0 || ix >= kW) continue;
            float v  = in[iy * kW + ix];
            float av = fabsf(v);
            float f0 = sf[ky * 7 + kx];
            float f1 = sf[49 + ky * 7 + kx];
            a0 += v * f0;  a1 += v * f1;
            w0 += av * f0; w1 += av * f1;
        }
    }
    long base = (long)b * 2 * kHW;
    gimg[base + p]        = a0 * kExtF;
    gimg[base + kHW + p]  = a1 * kExtF;
    gimgW[base + p]       = w0 * kExtF;
    gimgW[base + kHW + p] = w1 * kExtF;
}

// ---------------- snake evolution: one block per sample, one thread per node ----------------
__global__ void snake_kernel(const float* __restrict__ nodes, const float* __restrict__ widths,
                             const float* __restrict__ gimg, const float* __restrict__ gimgW,
                             float* __restrict__ xout, float* __restrict__ wout) {
    __shared__ float sy[kN], sx[kN], s2y[kN], s2x[kN];
    int b = blockIdx.x;
    int i = threadIdx.x;              // 0..95, exactly 3 wave32s

    const float* g0 = gimg + (long)b * 2 * kHW;
    const float* g1 = g0 + kHW;

    sy[i] = nodes[b * kN * 2 + i * 2 + 0];
    sx[i] = nodes[b * kN * 2 + i * 2 + 1];
    __syncthreads();

    int im = i > 0 ? i - 1 : 0;
    int ip = i < kN - 1 ? i + 1 : kN - 1;

    for (int s = 0; s < kSteps; s++) {
        float cy = sy[i], cx = sx[i];
        float d2a = sy[im] - 2.f * cy + sy[ip];
        float d2b = sx[im] - 2.f * cx + sx[ip];
        float fy = bilerp(g0, cy, cx);
        float fx = bilerp(g1, cy, cx);
        s2y[i] = d2a; s2x[i] = d2b;
        __syncthreads();
        float d4a = s2y[im] - 2.f * d2a + s2y[ip];
        float d4b = s2x[im] - 2.f * d2b + s2x[ip];
        float nyv = cy + kStep * (kAlpha * d2a - kBeta * d4a + fy);
        float nxv = cx + kStep * (kAlpha * d2b - kBeta * d4b + fx);
        __syncthreads();
        sy[i] = nyv; sx[i] = nxv;
        __syncthreads();
    }

    // tangent -> normal
    float ty, tx;
    if (i == 0)            { ty = sy[1] - sy[0];           tx = sx[1] - sx[0]; }
    else if (i == kN - 1)  { ty = sy[kN-1] - sy[kN-2];     tx = sx[kN-1] - sx[kN-2]; }
    else                   { ty = 0.5f * (sy[i+1] - sy[i-1]); tx = 0.5f * (sx[i+1] - sx[i-1]); }
    float n0 = -tx, n1 = ty;
    float nrm = sqrtf(n0 * n0 + n1 * n1) + 1e-6f;
    n0 /= nrm; n1 /= nrm;

    const float* gw0 = gimgW + (long)b * 2 * kHW;
    const float* gw1 = gw0 + kHW;
    float w  = widths[b * kN + i];
    float cy = sy[i], cx = sx[i];
    for (int s = 0; s < kStepsW; s++) {
        float pyv = cy + w * n0, pxv = cx + w * n1;
        float myv = cy - w * n0, mxv = cx - w * n1;
        float gp = bilerp(gw0, pyv, pxv) * n0 + bilerp(gw1, pyv, pxv) * n1;
        float gm = bilerp(gw0, myv, mxv) * n0 + bilerp(gw1, myv, mxv) * n1;
        w = fmaxf(w + kStep * 0.5f * (gp - gm), 0.5f);
    }

    xout[b * kN * 2 + i * 2 + 0] = cy;
    xout[b * kN * 2 + i * 2 + 1] = cx;
    wout[b * kN + i] = w;
}

// ---------------- render + loss partials: WMMA f32 16x16x4 for P.ab and P.a ----------------
// grid (kHW/128, kB), 256 threads = 8 waves; each wave: 16 pixels x 96 segments.
__global__ void render_kernel(const float* __restrict__ pred, const float* __restrict__ xs,
                              const float* __restrict__ wnode, float* __restrict__ partials) {
    __shared__ float s_ay[96], s_ax[96], s_aby[96], s_abx[96];
    __shared__ float s_den[96], s_inv[96], s_adab[96], s_a2[96], s_hw[96];
    __shared__ float s_wavesum[8];

    int b = blockIdx.y;
    int tid = threadIdx.x;

    if (tid < 96) {
        int i = tid > 94 ? 94 : tid;   // slot 95 duplicates segment 94 (min is idempotent)
        float ay = xs[b * kN * 2 + i * 2 + 0];
        float ax = xs[b * kN * 2 + i * 2 + 1];
        float by = xs[b * kN * 2 + (i + 1) * 2 + 0];
        float bx = xs[b * kN * 2 + (i + 1) * 2 + 1];
        float aby = by - ay, abx = bx - ax;
        float den = aby * aby + abx * abx + 1e-8f;
        s_ay[tid] = ay; s_ax[tid] = ax; s_aby[tid] = aby; s_abx[tid] = abx;
        s_den[tid] = den; s_inv[tid] = 1.f / den;
        s_adab[tid] = ay * aby + ax * abx;
        s_a2[tid]  = ay * ay + ax * ax;
        s_hw[tid]  = 0.5f * (wnode[b * kN + i] + wnode[b * kN + i + 1]);
    }
    __syncthreads();

    int lane = tid & 31;
    int wid  = tid >> 5;
    int half = lane >> 4;        // K-half for A/B operands, M-half for C/D
    int col  = lane & 15;        // M row (A) / N column (B, C/D)
    int tilebase = (blockIdx.x * 8 + wid) * 16;

    // A operand: 16x4, row M = pixel, K = (py, px, 0, 0).
    // Lanes 0-15 hold K=0,1 in the two VGPRs; lanes 16-31 hold K=2,3 (zeros).
    int pmA = tilebase + col;
    int pyA = pmA / kW;
    int pxA = pmA - pyA * kW;
    v2f aop;
    aop.x = half ? 0.f : (float)pyA;
    aop.y = half ? 0.f : (float)pxA;

    // per-(lane,VGPR) pixel ids and |P|^2: VGPR r -> M = r + 8*half
    float p2[8];
    int   pixr[8];
    #pragma unroll
    for (int r = 0; r < 8; r++) {
        int pix = tilebase + r + 8 * half;
        pixr[r] = pix;
        int py = pix / kW;
        int px = pix - py * kW;
        p2[r] = (float)(py * py + px * px);
    }

    float minv[8];
    #pragma unroll
    for (int r = 0; r < 8; r++) minv[r] = 3.0e38f;

    for (int st = 0; st < 6; st++) {
        int seg = st * 16 + col;
        v2f bab, ba;
        bab.x = half ? 0.f : s_aby[seg];
        bab.y = half ? 0.f : s_abx[seg];
        ba.x  = half ? 0.f : s_ay[seg];
        ba.y  = half ? 0.f : s_ax[seg];

        v8f cab = {};
        v8f ca  = {};
        // D = A x B + C   (C = 0):  P.ab and P.a for a 16x16 tile
        cab = __builtin_amdgcn_wmma_f32_16x16x4_f32(false, aop, false, bab, (short)0, cab, false, false);
        ca  = __builtin_amdgcn_wmma_f32_16x16x4_f32(false, aop, false, ba,  (short)0, ca,  false, false);

        float den  = s_den[seg];
        float invd = s_inv[seg];
        float adab = s_adab[seg];
        float a2v  = s_a2[seg];
        float hwv  = s_hw[seg];
        #pragma unroll
        for (int r = 0; r < 8; r++) {
            float dotpa = cab[r] - adab;
            float t = dotpa * invd;
            t = t < 0.f ? 0.f : (t > 1.f ? 1.f : t);
            float pa2 = p2[r] - 2.f * ca[r] + a2v;
            float dd  = pa2 - 2.f * t * dotpa + t * t * den;
            float d   = sqrtf(fmaxf(dd, 0.f));
            float v   = fmaxf(d - hwv, 0.f);
            minv[r]   = fminf(minv[r], v);
        }
    }

    // min over the 16 segment-columns in each half, then squared error
    float local = 0.f;
    const float* pb = pred + (long)b * kHW;
    #pragma unroll
    for (int r = 0; r < 8; r++) {
        float v = minv[r];
        v = fminf(v, __shfl_xor(v, 1, 16));
        v = fminf(v, __shfl_xor(v, 2, 16));
        v = fminf(v, __shfl_xor(v, 4, 16));
        v = fminf(v, __shfl_xor(v, 8, 16));
        if (col == 0) {
            float dm = fminf(v, kDmax);
            float e  = pb[pixr[r]] - dm;
            local += e * e;
        }
    }

    // wave then block reduction of the loss partial
    local += __shfl_xor(local, 16, 32);
    local += __shfl_xor(local, 8, 32);
    local += __shfl_xor(local, 4, 32);
    local += __shfl_xor(local, 2, 32);
    local += __shfl_xor(local, 1, 32);
    if (lane == 0) s_wavesum[wid] = local;
    __syncthreads();
    if (tid == 0) {
        float s = 0.f;
        #pragma unroll
        for (int k = 0; k < 8; k++) s += s_wavesum[k];
        partials[b * gridDim.x + blockIdx.x] = s;
    }
}

// ---------------- final deterministic reduction ----------------
__global__ void final_reduce(const float* __restrict__ partials, int n, float* __restrict__ out) {
    __shared__ double sd[256];
    double s = 0.0;
    for (int i = threadIdx.x; i < n; i += 256) s += (double)partials[i];
    sd[threadIdx.x] = s;
    __syncthreads();
    for (int o = 128; o > 0; o >>= 1) {
        if (threadIdx.x < o) sd[threadIdx.x] += sd[threadIdx.x + o];
        __syncthreads();
    }
    if (threadIdx.x == 0) out[0] = (float)(sd[0] / (double)((long)kB * kHW));
}

extern "C" void kernel_launch(void* const* d_in, const int* in_sizes, int n_in,
                              void* d_out, int out_size, void* d_ws, size_t ws_size,
                              hipStream_t stream) {
    const float* pred   = (const float*)d_in[0];   // (B,1,H,W)
    const float* nodes  = (const float*)d_in[1];   // (B,N,2)
    const float* widths = (const float*)d_in[2];   // (B,N)
    const float* fltr   = (const float*)d_in[3];   // (2,1,7,7)

    float* ws     = (float*)d_ws;
    float* gimg   = ws;                                   // B*2*HW
    float* gimgW  = gimg  + (size_t)kB * 2 * kHW;         // B*2*HW
    float* xout   = gimgW + (size_t)kB * 2 * kHW;         // B*N*2
    float* wout   = xout  + (size_t)kB * kN * 2;          // B*N
    float* parts  = wout  + (size_t)kB * kN;              // B*(HW/128) = 9216

    conv_kernel<<<(kB * kHW + 255) / 256, 256, 0, stream>>>(pred, fltr, gimg, gimgW);
    snake_kernel<<<kB, kN, 0, stream>>>(nodes, widths, gimg, gimgW, xout, wout);

    dim3 rgrid(kHW / 128, kB);
    render_kernel<<<rgrid, 256, 0, stream>>>(pred, xout, wout, parts);

    final_reduce<<<1, 256, 0, stream>>>(parts, kB * (kHW / 128), (float*)d_out);
}